// ROIPooling_1623497637911
// MI455X (gfx1250) — compile-verified
//
#include <hip/hip_runtime.h>
#include <cstdint>

// ROI max-pool: feat [B,H,W,C]=[2,50,50,512] f32, rois [B,N,4]=[2,128,4],
// out [B,N,7,7,512] f32.
// One wave32 per output bin, one wave per block so bin = blockIdx.x is scalar:
// bin decode -> SALU, ROI fetch -> s_load_b128, loop bounds -> scalar.
// Channels (512 f32 = 2KB/pixel) staged via gfx1250 async global->LDS DMA
// (global_load_async_to_lds_b128 / ASYNCcnt), 4-buffer depth-2 pipeline.

#define POOL 7
static constexpr int Bb = 2, Nn = 128, Hh = 50, Ww = 50, Cc = 512;
static constexpr float INV_STRIDE = 1.0f / 16.0f;

__device__ __forceinline__ uint32_t lds_off_of(const void* p) {
    // AS(3) pointers flatten to {shared_aperture_hi32, lds_offset}; low 32 bits = LDS byte addr
    return (uint32_t)(uintptr_t)p;
}

// Issue 4 async B128 loads: 32 lanes x 16B x 4 = 2048B = one pixel's 512 channels -> LDS
__device__ __forceinline__ void issue_pixel_async(unsigned long long pix_gbase,
                                                  uint32_t lds_base, int lane) {
#pragma unroll
    for (int k = 0; k < 4; ++k) {
        unsigned long long g = pix_gbase + (unsigned)((k * 32 + lane) * 16);
        uint32_t l = lds_base + (uint32_t)((k * 32 + lane) * 16);
        asm volatile("global_load_async_to_lds_b128 %0, %1, off"
                     :: "v"(l), "v"(g)
                     : "memory");
    }
}

__device__ __forceinline__ void consume_max(const float4* bufp, int lane, float4 (&acc)[4]) {
#pragma unroll
    for (int k = 0; k < 4; ++k) {
        float4 v = bufp[k * 32 + lane];
        acc[k].x = fmaxf(acc[k].x, v.x);
        acc[k].y = fmaxf(acc[k].y, v.y);
        acc[k].z = fmaxf(acc[k].z, v.z);
        acc[k].w = fmaxf(acc[k].w, v.w);
    }
}

__global__ __launch_bounds__(32) void roi_maxpool_kernel(const float* __restrict__ feat,
                                                         const float* __restrict__ rois,
                                                         float* __restrict__ out) {
    __shared__ float4 smem[4][128];      // 4 buffers x 2KB = 8KB (depth-2 pipeline)

    const int lane = threadIdx.x;        // 0..31
    const int bin  = blockIdx.x;         // scalar: ((b*N + n)*7 + ph)*7 + pw

    // --- all-scalar bin decode ---
    int b   = bin / (Nn * POOL * POOL);
    int rem = bin - b * (Nn * POOL * POOL);
    int n   = rem / (POOL * POOL);
    int q   = rem - n * (POOL * POOL);
    int ph  = q / POOL;
    int pw  = q - ph * POOL;

    // uniform address -> scalar load through constant cache
    const float4 r = reinterpret_cast<const float4*>(rois)[b * Nn + n];
    float x1 = fminf(fmaxf(r.x * INV_STRIDE, 0.0f), (float)(Ww - 1));
    float y1 = fminf(fmaxf(r.y * INV_STRIDE, 0.0f), (float)(Hh - 1));
    float x2 = fminf(fmaxf(r.z * INV_STRIDE, x1 + 1.0f), (float)Ww);
    float y2 = fminf(fmaxf(r.w * INV_STRIDE, y1 + 1.0f), (float)Hh);
    float bw = (x2 - x1) * (1.0f / POOL);
    float bh = (y2 - y1) * (1.0f / POOL);

    int sx = max((int)floorf(x1 + (float)pw * bw), 0);
    int ex = min((int)ceilf(x1 + (float)(pw + 1) * bw), Ww);
    int sy = max((int)floorf(y1 + (float)ph * bh), 0);
    int ey = min((int)ceilf(y1 + (float)(ph + 1) * bh), Hh);

    const int nx = ex - sx;
    const int ny = ey - sy;
    const int npix = (nx > 0 && ny > 0) ? nx * ny : 0;

    float4 acc[4];
#pragma unroll
    for (int k = 0; k < 4; ++k)
        acc[k] = make_float4(-__builtin_inff(), -__builtin_inff(),
                             -__builtin_inff(), -__builtin_inff());

    const uint32_t lbase0 = lds_off_of(&smem[0][0]);

    if (npix > 0) {
        const unsigned long long row_stride = (unsigned long long)(Ww * Cc * 4); // 102400 B
        const unsigned long long pix_stride = (unsigned long long)(Cc * 4);      // 2048 B
        const unsigned long long row_adv    = row_stride - (unsigned long long)nx * pix_stride;

        // row-major walk over the bin region, flattened with incremental addresses
        unsigned long long gnext =
            (unsigned long long)(uintptr_t)feat +
            (unsigned long long)((b * Hh + sy) * Ww + sx) * pix_stride;
        unsigned long long grow_end = gnext + (unsigned long long)nx * pix_stride;

        // prologue: p0 (and p1 if it exists) in flight
        issue_pixel_async(gnext, lbase0, lane);
        gnext += pix_stride;
        if (gnext == grow_end) { gnext += row_adv; grow_end = gnext + (unsigned long long)nx * pix_stride; }
        if (npix > 1) {
            issue_pixel_async(gnext, lbase0 + 2048u, lane);
            gnext += pix_stride;
            if (gnext == grow_end) { gnext += row_adv; grow_end = gnext + (unsigned long long)nx * pix_stride; }
        }

        // steady state: issue p(i+2), wait until p(i) complete, consume p(i)
        for (int i = 0; i < npix - 2; ++i) {
            // WAR guard: LDS reads of buffer (i+2)&3 (consumed 2 iters ago) must have
            // drained; DScnt and ASYNCcnt are mutually unordered.
            asm volatile("s_wait_dscnt 0x0" ::: "memory");
            issue_pixel_async(gnext, lbase0 + 2048u * (uint32_t)((i + 2) & 3), lane);
            gnext += pix_stride;
            if (gnext == grow_end) { gnext += row_adv; grow_end = gnext + (unsigned long long)nx * pix_stride; }
            // pixels i, i+1, i+2 in flight = 12 async ops; async loads complete in
            // order, so <=8 outstanding means pixel i is fully in LDS.
            asm volatile("s_wait_asynccnt 0x8" ::: "memory");
            consume_max(&smem[i & 3][0], lane, acc);
        }

        // tail: up to two pixels still in flight
        if (npix >= 2) {
            asm volatile("s_wait_asynccnt 0x4" ::: "memory");
            consume_max(&smem[(npix - 2) & 3][0], lane, acc);
        }
        asm volatile("s_wait_asynccnt 0x0" ::: "memory");
        consume_max(&smem[(npix - 1) & 3][0], lane, acc);
    } else {
#pragma unroll
        for (int k = 0; k < 4; ++k) acc[k] = make_float4(0.f, 0.f, 0.f, 0.f);
    }

    float4* o4 = reinterpret_cast<float4*>(out);
    const long long obase = (long long)bin * (Cc / 4);   // float4 index
#pragma unroll
    for (int k = 0; k < 4; ++k)
        o4[obase + k * 32 + lane] = acc[k];
}

extern "C" void kernel_launch(void* const* d_in, const int* in_sizes, int n_in,
                              void* d_out, int out_size, void* d_ws, size_t ws_size,
                              hipStream_t stream) {
    (void)in_sizes; (void)n_in; (void)d_ws; (void)ws_size; (void)out_size;
    const float* feat = (const float*)d_in[0];   // [2,50,50,512]
    const float* rois = (const float*)d_in[1];   // [2,128,4]
    float* out = (float*)d_out;                  // [2,128,7,7,512]

    const int bins = Bb * Nn * POOL * POOL;      // 12544 = one wave32 (one block) per bin
    roi_maxpool_kernel<<<bins, 32, 0, stream>>>(feat, rois, out);
}